// LKA_dynamic_44487271252034
// MI455X (gfx1250) — compile-verified
//
#include <hip/hip_runtime.h>
#include <math.h>

typedef __attribute__((ext_vector_type(2))) float v2f;
typedef __attribute__((ext_vector_type(8))) float v8f;

#define DIMC 64
#define HW   16384   // 128*128

__device__ __forceinline__ float gelu_exact(float v) {
    return 0.5f * v * (1.0f + erff(v * 0.70710678118654752f));
}

// ---------------------------------------------------------------------------
// Plane mean: one workgroup per (b,c) plane of 128x128 floats.
// ---------------------------------------------------------------------------
__global__ __launch_bounds__(256) void pool_kernel(const float* __restrict__ in,
                                                   float* __restrict__ pooled) {
    __shared__ float red[256];
    const int bc = blockIdx.x, tid = threadIdx.x;
    const float4* p4 = (const float4*)(in + (size_t)bc * HW);
    float s = 0.0f;
#pragma unroll
    for (int i = 0; i < 16; ++i) {
        float4 v = p4[tid + 256 * i];
        s += v.x + v.y + v.z + v.w;
    }
    red[tid] = s;
    __syncthreads();
    for (int off = 128; off > 0; off >>= 1) {
        if (tid < off) red[tid] += red[tid + off];
        __syncthreads();
    }
    if (tid == 0) pooled[bc] = red[0] * (1.0f / 16384.0f);
}

// ---------------------------------------------------------------------------
// Routing MLP + expert-kernel mix. One block (64 threads) per sample.
// h = LeakyReLU(pooled @ fc1w^T + fc1b); r = sigmoid(h @ fc2w^T + fc2b)
// kmix[b,c,j] = sum_e r[e] * w[e,c,j]
// ---------------------------------------------------------------------------
__global__ __launch_bounds__(64) void route_kernel(const float* __restrict__ pooled,
                                                   const float* __restrict__ fc1w,
                                                   const float* __restrict__ fc1b,
                                                   const float* __restrict__ fc2w,
                                                   const float* __restrict__ fc2b,
                                                   const float* __restrict__ w,
                                                   float* __restrict__ kmix,
                                                   int KK) {
    __shared__ float ps[DIMC], hs[DIMC], rs[3];
    const int b = blockIdx.x, t = threadIdx.x;
    ps[t] = pooled[b * DIMC + t];
    __syncthreads();
    float a = fc1b[t];
    for (int k = 0; k < DIMC; ++k) a = fmaf(fc1w[t * DIMC + k], ps[k], a);
    hs[t] = (a >= 0.0f) ? a : 0.1f * a;
    __syncthreads();
    if (t < 3) {
        float a2 = fc2b[t];
        for (int k = 0; k < DIMC; ++k) a2 = fmaf(fc2w[t * DIMC + k], hs[k], a2);
        rs[t] = 1.0f / (1.0f + expf(-a2));
    }
    __syncthreads();
    const int n = DIMC * KK;
    const float r0 = rs[0], r1 = rs[1], r2 = rs[2];
    for (int j = t; j < n; j += 64)
        kmix[(size_t)b * n + j] = r0 * w[j] + r1 * w[n + j] + r2 * w[2 * n + j];
}

// ---------------------------------------------------------------------------
// Depthwise conv (per-sample mixed kernel) + bias + exact GELU.
// One workgroup per (b,c) plane; full plane staged in 64 KB LDS (CDNA5 WGP
// has 320 KB, so 2+ blocks/WGP). Optionally emits the plane mean of the
// activated output (pooled for the next routing stage) — no atomics needed
// since exactly one WG owns each plane. Safe to run in-place (out == in):
// the full input plane is in LDS behind a barrier before any global write.
// ---------------------------------------------------------------------------
template <int KS, int DIL, int PAD, bool POOL>
__global__ __launch_bounds__(256) void dwconv_kernel(const float* __restrict__ in,
                                                     float* __restrict__ out,
                                                     const float* __restrict__ kmix,
                                                     const float* __restrict__ bias,
                                                     float* __restrict__ pooled_out) {
    __shared__ float xs[HW];
    __shared__ float red[256];
    const int bc = blockIdx.x, tid = threadIdx.x;
    const int c = bc & (DIMC - 1);

    // Stage the whole 128x128 plane into LDS with float4 loads.
    const float4* in4 = (const float4*)(in + (size_t)bc * HW);
    float4* xs4 = (float4*)xs;
#pragma unroll
    for (int i = 0; i < 16; ++i) xs4[tid + 256 * i] = in4[tid + 256 * i];

    float wk[KS * KS];
    const float* kb = kmix + (size_t)bc * (KS * KS);
#pragma unroll
    for (int j = 0; j < KS * KS; ++j) wk[j] = kb[j];
    const float bch = bias[c];
    __syncthreads();

    float* ob = out + (size_t)bc * HW;
    float lsum = 0.0f;
    for (int i = 0; i < 64; ++i) {
        const int p = i * 256 + tid;
        const int y = p >> 7, xq = p & 127;
        float acc = 0.0f;
        if (y >= PAD && y < 128 - PAD && xq >= PAD && xq < 128 - PAD) {
#pragma unroll
            for (int ty = 0; ty < KS; ++ty)
#pragma unroll
                for (int tx = 0; tx < KS; ++tx)
                    acc = fmaf(wk[ty * KS + tx],
                               xs[(y + ty * DIL - PAD) * 128 + (xq + tx * DIL - PAD)], acc);
        } else {
#pragma unroll
            for (int ty = 0; ty < KS; ++ty) {
                const int yy = y + ty * DIL - PAD;
                if ((unsigned)yy < 128u) {
#pragma unroll
                    for (int tx = 0; tx < KS; ++tx) {
                        const int xx = xq + tx * DIL - PAD;
                        if ((unsigned)xx < 128u)
                            acc = fmaf(wk[ty * KS + tx], xs[yy * 128 + xx], acc);
                    }
                }
            }
        }
        const float v = gelu_exact(acc + bch);
        ob[p] = v;
        lsum += v;
    }
    if (POOL) {
        __syncthreads();
        red[tid] = lsum;
        __syncthreads();
        for (int off = 128; off > 0; off >>= 1) {
            if (tid < off) red[tid] += red[tid + off];
            __syncthreads();
        }
        if (tid == 0) pooled_out[bc] = red[0] * (1.0f / 16384.0f);
    }
}

// ---------------------------------------------------------------------------
// 1x1 conv as a GEMM via fp32 WMMA (V_WMMA_F32_16X16X4_F32), fused bias and
// residual multiply:  out[b,o,p] = (sum_c t[b,c,p]*w[o,c] + bias[o]) * x[b,o,p].
// Each wave owns a 16-pixel x 16-outchannel tile; K=64 done as 16 chained
// x4 WMMAs. fp32 A layout: lane = {g=lane/16 -> K in {2g,2g+1}, M=lane%16};
// B mirrors with N; C/D v8f element i -> (M = i + 8g, N = lane%16).
// ---------------------------------------------------------------------------
__global__ __launch_bounds__(256) void pw_wmma_kernel(const float* __restrict__ t,
                                                      const float* __restrict__ w,
                                                      const float* __restrict__ bias,
                                                      const float* __restrict__ x,
                                                      float* __restrict__ out) {
    const int tid  = threadIdx.x;
    const int wave = tid >> 5, lane = tid & 31;
    const int g = lane >> 4, r = lane & 15;
    const int ntile  = wave & 3;
    const int Mtile  = blockIdx.x * 2 + (wave >> 2);
    const int P0 = Mtile << 4;              // global flat pixel base (B*HW space)
    const int b  = P0 >> 14;                // HW == 16384
    const int p0 = P0 & (HW - 1);           // 16-aligned, never crosses a batch
    const int n0 = ntile << 4;

    const float* tb = t + ((size_t)b * DIMC) * HW;
    v8f acc = {};
#pragma unroll
    for (int k0 = 0; k0 < DIMC; k0 += 4) {
        const int ch = k0 + 2 * g;
        v2f a, bw;
        const float* ta = tb + ((size_t)ch * HW) + p0 + r;
        a.x = ta[0];
        a.y = ta[HW];
        const float* wa = w + (n0 + r) * DIMC + ch;
        bw.x = wa[0];
        bw.y = wa[1];
        acc = __builtin_amdgcn_wmma_f32_16x16x4_f32(false, a, false, bw,
                                                    (short)0, acc, false, false);
    }
    const int o = n0 + r;
    const float bo = bias[o];
    const size_t off = ((size_t)(b * DIMC + o)) * HW + p0;
    const float* xb = x + off;
    float* ob = out + off;
#pragma unroll
    for (int i = 0; i < 8; ++i) {
        const int m = i + 8 * g;
        ob[m] = (acc[i] + bo) * xb[m];
    }
}

// ---------------------------------------------------------------------------
extern "C" void kernel_launch(void* const* d_in, const int* in_sizes, int n_in,
                              void* d_out, int out_size, void* d_ws, size_t ws_size,
                              hipStream_t stream) {
    const float* x      = (const float*)d_in[0];
    const float* w0     = (const float*)d_in[1];
    const float* b0     = (const float*)d_in[2];
    const float* fc1w0  = (const float*)d_in[3];
    const float* fc1b0  = (const float*)d_in[4];
    const float* fc2w0  = (const float*)d_in[5];
    const float* fc2b0  = (const float*)d_in[6];
    const float* w1     = (const float*)d_in[7];
    const float* b1     = (const float*)d_in[8];
    const float* fc1w1  = (const float*)d_in[9];
    const float* fc1b1  = (const float*)d_in[10];
    const float* fc2w1  = (const float*)d_in[11];
    const float* fc2b1  = (const float*)d_in[12];
    const float* conv1w = (const float*)d_in[13];
    const float* conv1b = (const float*)d_in[14];
    float* out = (float*)d_out;

    float* ws      = (float*)d_ws;
    float* pooled0 = ws;                 // 2048
    float* pooled1 = ws + 2048;          // 2048
    float* k0m     = ws + 4096;          // 32*64*25 = 51200
    float* k1m     = ws + 55296;         // 32*64*49 = 100352
    float* t       = ws + 155648;        // 32*64*16384 = 33554432

    const int BC = 32 * DIMC;            // 2048 planes

    pool_kernel<<<BC, 256, 0, stream>>>(x, pooled0);
    route_kernel<<<32, 64, 0, stream>>>(pooled0, fc1w0, fc1b0, fc2w0, fc2b0,
                                        w0, k0m, 25);
    dwconv_kernel<5, 1, 2, true><<<BC, 256, 0, stream>>>(x, t, k0m, b0, pooled1);
    route_kernel<<<32, 64, 0, stream>>>(pooled1, fc1w1, fc1b1, fc2w1, fc2b1,
                                        w1, k1m, 49);
    dwconv_kernel<7, 3, 9, false><<<BC, 256, 0, stream>>>(t, t, k1m, b1, nullptr);
    pw_wmma_kernel<<<32 * HW / 32, 256, 0, stream>>>(t, conv1w, conv1b, x, out);
}